// MambaBlock_47588237639818
// MI455X (gfx1250) — compile-verified
//
#include <hip/hip_runtime.h>
#include <math.h>
#include <stdint.h>

// ---------------------------------------------------------------------------
// Mamba block for MI455X (gfx1250, wave32). fp32 end-to-end using
// V_WMMA_F32_16X16X4_F32 for the two large GEMMs and for the chunked
// parallel scan's 16x16 matrix-product recurrence. Weight tiles are staged
// into LDS with CDNA5 async global->LDS loads (ASYNCcnt) and double-buffered.
// ---------------------------------------------------------------------------

typedef float v2f __attribute__((ext_vector_type(2)));
typedef float v8f __attribute__((ext_vector_type(8)));

#define BATCH   2
#define SEQ     4096
#define DM      1024
#define DI      2048
#define NPROJ   4096          // 2*DI
#define MTOT    (BATCH*SEQ)   // 8192 token rows
#define DS      16
#define LC      64            // scan chunk length
#define NCHUNK  (MTOT/LC)     // 128
#define LN_EPS  1e-5f

#define BLK 64
#define KC  32                // K slab per LDS tile
#define KP  36                // padded LDS row stride (floats): 144B, 16B-aligned,
                              // 36*idx mod 64 -> 16 distinct banks (conflict-free)

__device__ __forceinline__ v8f wmma4(v2f a, v2f b, v8f c) {
  // D = A(16x4,f32) x B(4x16,f32) + C(16x16,f32)
  return __builtin_amdgcn_wmma_f32_16x16x4_f32(false, a, false, b, (short)0, c,
                                               false, false);
}

// CDNA5 async copy: global memory -> LDS, 16B per lane, tracked by ASYNCcnt.
__device__ __forceinline__ void async_load_b128(void* lds_dst, const void* gaddr) {
  const uint32_t lds_off = (uint32_t)(uintptr_t)lds_dst;  // low 32b of LDS-aperture addr
  asm volatile("global_load_async_to_lds_b128 %0, %1, off"
               :: "v"(lds_off), "v"(gaddr) : "memory");
}
__device__ __forceinline__ void wait_async0() {
  asm volatile("s_wait_asynccnt 0" ::: "memory");
}

__device__ __forceinline__ float wave_red32(float v) {
#pragma unroll
  for (int o = 16; o > 0; o >>= 1) v += __shfl_down(v, o, 32);
  return v;
}

__device__ __forceinline__ float silu_f(float v) {
  return v * (1.0f / (1.0f + __expf(-v)));
}

// ---------------------------------------------------------------------------
// Kernel 1: per-row LayerNorm statistics (mean, rsqrt(var+eps))
// ---------------------------------------------------------------------------
__global__ __launch_bounds__(256) void ln_stats_k(const float* __restrict__ x,
                                                  float* __restrict__ mu,
                                                  float* __restrict__ rs) {
  __shared__ float red[8];
  const int m = blockIdx.x, tid = threadIdx.x;
  const int lane = tid & 31, wave = tid >> 5;
  const float4 v = *(const float4*)(x + (size_t)m * DM + tid * 4);
  float s = v.x + v.y + v.z + v.w;
  s = wave_red32(s);
  if (lane == 0) red[wave] = s;
  __syncthreads();
  float tot = 0.f;
#pragma unroll
  for (int w = 0; w < 8; ++w) tot += red[w];
  const float mean = tot * (1.0f / DM);
  float dx = v.x - mean, dy = v.y - mean, dz = v.z - mean, dw = v.w - mean;
  float s2 = dx * dx + dy * dy + dz * dz + dw * dw;
  __syncthreads();
  s2 = wave_red32(s2);
  if (lane == 0) red[wave] = s2;
  __syncthreads();
  if (tid == 0) {
    float t2 = 0.f;
#pragma unroll
    for (int w = 0; w < 8; ++w) t2 += red[w];
    mu[m] = mean;
    rs[m] = rsqrtf(t2 * (1.0f / DM) + LN_EPS);
  }
}

// ---------------------------------------------------------------------------
// Kernel 2: GEMM1  Y1[8192,4096] = LayerNorm(x)[8192,1024] @ Win^T[1024,4096]
// LN fused on A-tile prologue; B tiles via async global->LDS; double-buffered.
// ---------------------------------------------------------------------------
__global__ __launch_bounds__(256) void gemm1_ln_k(const float* __restrict__ x,
                                                  const float* __restrict__ mu,
                                                  const float* __restrict__ rs,
                                                  const float* __restrict__ gamma,
                                                  const float* __restrict__ beta,
                                                  const float* __restrict__ W,
                                                  float* __restrict__ Y) {
  __shared__ float As[2][BLK][KP];
  __shared__ float Bs[2][BLK][KP];
  const int tid  = threadIdx.x;
  const int m0   = blockIdx.y * BLK;
  const int n0   = blockIdx.x * BLK;
  const int lane = tid & 31, wave = tid >> 5;
  const int half = lane >> 4, idx = lane & 15;
  const int wm   = (wave & 3) * 16;   // wave's M sub-tile
  const int wn   = (wave >> 2) * 32;  // wave's N sub-tile pair
  const int lr   = tid >> 2;          // 0..63 load row
  const int c0   = (tid & 3) * 4;     // this thread's chunk cols: c0, c0+16
  const float mu_m = mu[m0 + lr];
  const float rs_m = rs[m0 + lr];

  auto loadB = [&](int buf, int k0) {
    const float* src = W + (size_t)(n0 + lr) * DM + k0 + c0;
    async_load_b128(&Bs[buf][lr][c0],      src);
    async_load_b128(&Bs[buf][lr][c0 + 16], src + 16);
  };
  auto loadA = [&](int buf, int k0) {
    const float* src = x + (size_t)(m0 + lr) * DM + k0 + c0;
    const float4 x0 = *(const float4*)(src);
    const float4 x1 = *(const float4*)(src + 16);
    const float4 g0 = *(const float4*)(gamma + k0 + c0);
    const float4 g1 = *(const float4*)(gamma + k0 + c0 + 16);
    const float4 b0 = *(const float4*)(beta + k0 + c0);
    const float4 b1 = *(const float4*)(beta + k0 + c0 + 16);
    float4 a0, a1;
    a0.x = (x0.x - mu_m) * rs_m * g0.x + b0.x;
    a0.y = (x0.y - mu_m) * rs_m * g0.y + b0.y;
    a0.z = (x0.z - mu_m) * rs_m * g0.z + b0.z;
    a0.w = (x0.w - mu_m) * rs_m * g0.w + b0.w;
    a1.x = (x1.x - mu_m) * rs_m * g1.x + b1.x;
    a1.y = (x1.y - mu_m) * rs_m * g1.y + b1.y;
    a1.z = (x1.z - mu_m) * rs_m * g1.z + b1.z;
    a1.w = (x1.w - mu_m) * rs_m * g1.w + b1.w;
    *(float4*)&As[buf][lr][c0]      = a0;
    *(float4*)&As[buf][lr][c0 + 16] = a1;
  };

  v8f acc0; v8f acc1;
#pragma unroll
  for (int q = 0; q < 8; ++q) { acc0[q] = 0.f; acc1[q] = 0.f; }

  loadB(0, 0);
  loadA(0, 0);
  wait_async0();
  __syncthreads();

  int buf = 0;
  for (int k0 = 0; k0 < DM; k0 += KC) {
    const int nb = buf ^ 1;
    const bool more = (k0 + KC < DM);
    if (more) loadB(nb, k0 + KC);
#pragma unroll
    for (int kk = 0; kk < KC; kk += 4) {
      const v2f a  = *(const v2f*)&As[buf][wm + idx][kk + 2 * half];
      const v2f b0 = *(const v2f*)&Bs[buf][wn + idx][kk + 2 * half];
      const v2f b1 = *(const v2f*)&Bs[buf][wn + 16 + idx][kk + 2 * half];
      acc0 = wmma4(a, b0, acc0);
      acc1 = wmma4(a, b1, acc1);
    }
    if (more) loadA(nb, k0 + KC);
    wait_async0();
    __syncthreads();
    buf = nb;
  }
#pragma unroll
  for (int r = 0; r < 8; ++r) {
    const int m = m0 + wm + 8 * half + r;
    Y[(size_t)m * NPROJ + n0 + wn + idx]      = acc0[r];
    Y[(size_t)m * NPROJ + n0 + wn + 16 + idx] = acc1[r];
  }
}

// ---------------------------------------------------------------------------
// Kernel 3: causal depthwise conv (K=4) + bias + SiLU -> xc; fused skinny
// projections dt (softplus), Bm[16], Cm[16] per token via block reduction.
// One block per token row m = b*SEQ + l.
// ---------------------------------------------------------------------------
__global__ __launch_bounds__(256) void conv_proj_k(const float* __restrict__ Y1,
                                                   const float* __restrict__ cw,
                                                   const float* __restrict__ cb,
                                                   const float* __restrict__ dtw,
                                                   const float* __restrict__ dtb,
                                                   const float* __restrict__ Bw,
                                                   const float* __restrict__ Cw,
                                                   float* __restrict__ xc,
                                                   float* __restrict__ dtv,
                                                   float* __restrict__ Bm,
                                                   float* __restrict__ Cm) {
  __shared__ float red[8 * 33];
  const int m = blockIdx.x;
  const int b = m >> 12;        // m / 4096
  const int l = m & 4095;
  const int tid = threadIdx.x;
  float dtp = 0.f, bp[DS], cp[DS];
#pragma unroll
  for (int n = 0; n < DS; ++n) { bp[n] = 0.f; cp[n] = 0.f; }

  for (int j = 0; j < 8; ++j) {
    const int c = tid + j * 256;
    float sum = cb[c];
#pragma unroll
    for (int tap = 0; tap < 4; ++tap) {
      const int lp = l - 3 + tap;
      if (lp >= 0)
        sum += Y1[((size_t)(b * SEQ + lp)) * NPROJ + c] * cw[c * 4 + tap];
    }
    const float v = silu_f(sum);
    xc[(size_t)m * DI + c] = v;
    dtp += v * dtw[c];
#pragma unroll
    for (int n = 0; n < DS; ++n) {
      bp[n] += v * Bw[n * DI + c];
      cp[n] += v * Cw[n * DI + c];
    }
  }
  const int lane = tid & 31, wave = tid >> 5;
  dtp = wave_red32(dtp);
#pragma unroll
  for (int n = 0; n < DS; ++n) { bp[n] = wave_red32(bp[n]); cp[n] = wave_red32(cp[n]); }
  if (lane == 0) {
    red[wave * 33 + 0] = dtp;
#pragma unroll
    for (int n = 0; n < DS; ++n) {
      red[wave * 33 + 1 + n]  = bp[n];
      red[wave * 33 + 17 + n] = cp[n];
    }
  }
  __syncthreads();
  if (tid < 33) {
    float s = 0.f;
#pragma unroll
    for (int w = 0; w < 8; ++w) s += red[w * 33 + tid];
    if (tid == 0) {
      const float z = s + dtb[0];
      dtv[m] = (z > 20.f) ? z : log1pf(__expf(z));   // softplus
    } else if (tid < 17) {
      Bm[(size_t)m * DS + (tid - 1)] = s;
    } else {
      Cm[(size_t)m * DS + (tid - 17)] = s;
    }
  }
}

// ---------------------------------------------------------------------------
// Kernel 4: chunked scan. One wave32 per 64-token chunk (128 chunks).
// Maintains P_t = dA_t * P_{t-1} via 4 chained WMMA f32 16x16x4, plus the
// local state h (zero-init) in lanes. Emits per-token g_t = P_t^T C_t and
// partial_t = C_t . h_loc, plus per-chunk aggregates (M_c, b_c).
// ---------------------------------------------------------------------------
__global__ __launch_bounds__(32) void scan_chunks_k(const float* __restrict__ dt,
                                                    const float* __restrict__ Bm,
                                                    const float* __restrict__ Cm,
                                                    const float* __restrict__ Apar,
                                                    float* __restrict__ g,
                                                    float* __restrict__ part,
                                                    float* __restrict__ Mc,
                                                    float* __restrict__ bc) {
  __shared__ float Ps[16][17];
  const int lane = threadIdx.x;
  const int half = lane >> 4, idx = lane & 15;
  const int cid = blockIdx.x;     // 0..127, tokens are contiguous: m0 = cid*LC
  const int m0 = cid * LC;

  // per-lane -exp(A_param) coefficients at this lane's 8 (row=idx, k) slots
  float na[8];
#pragma unroll
  for (int j = 0; j < 8; ++j) {
    const int k = (j >> 1) * 4 + 2 * half + (j & 1);
    na[j] = -__expf(Apar[idx * DS + k]);
  }
  // P = Identity
  for (int e = lane; e < 256; e += 32) Ps[e >> 4][e & 15] = ((e >> 4) == (e & 15)) ? 1.f : 0.f;
  float h = 0.f;                   // lane holds h[idx], replicated across halves
  v8f acc;
#pragma unroll
  for (int q = 0; q < 8; ++q) acc[q] = 0.f;

  for (int t = 0; t < LC; ++t) {
    const int m = m0 + t;
    const float dtvv = dt[m];
    const float Bv = Bm[(size_t)m * DS + idx];
    const float Cv = Cm[(size_t)m * DS + idx];
    float da[8];
#pragma unroll
    for (int j = 0; j < 8; ++j) da[j] = __expf(dtvv * na[j]);
    __syncthreads();
#pragma unroll
    for (int q = 0; q < 8; ++q) acc[q] = 0.f;
#pragma unroll
    for (int kk = 0; kk < 4; ++kk) {
      v2f a, bfr;
      a.x = da[2 * kk + 0];
      a.y = da[2 * kk + 1];
      bfr.x = Ps[kk * 4 + 2 * half + 0][idx];
      bfr.y = Ps[kk * 4 + 2 * half + 1][idx];
      acc = wmma4(a, bfr, acc);
    }
    // h' = dA h + dt*B   (each lane covers its 8 k's, halves combined)
    float hp = 0.f;
#pragma unroll
    for (int j = 0; j < 8; ++j) {
      const int k = (j >> 1) * 4 + 2 * half + (j & 1);
      hp += da[j] * __shfl(h, k, 32);
    }
    hp += __shfl_xor(hp, 16, 32);
    h = hp + dtvv * Bv;
    // partial_t = C . h
    float r = h * Cv;
#pragma unroll
    for (int o = 8; o > 0; o >>= 1) r += __shfl_xor(r, o, 32);
    if (lane == 0) part[m] = r;
    // g_t = P'^T C : lane holds column idx, rows 8*half..8*half+7
    float gp = 0.f;
#pragma unroll
    for (int q = 0; q < 8; ++q) gp += acc[q] * __shfl(Cv, 8 * half + q, 32);
    gp += __shfl_xor(gp, 16, 32);
    if (half == 0) g[(size_t)m * DS + idx] = gp;
    __syncthreads();
#pragma unroll
    for (int q = 0; q < 8; ++q) Ps[8 * half + q][idx] = acc[q];
  }
  // chunk aggregates
#pragma unroll
  for (int q = 0; q < 8; ++q)
    Mc[(size_t)cid * 256 + (8 * half + q) * DS + idx] = acc[q];
  if (half == 0) bc[(size_t)cid * DS + idx] = h;
}

// ---------------------------------------------------------------------------
// Kernel 5: sequential combine of chunk aggregates. One wave per batch.
// hs[cid] = state at chunk start.
// ---------------------------------------------------------------------------
__global__ __launch_bounds__(64) void combine_k(const float* __restrict__ Mc,
                                                const float* __restrict__ bc,
                                                float* __restrict__ hs) {
  const int tid = threadIdx.x;
  const int b = tid >> 5;
  const int lane = tid & 31;
  const int half = lane >> 4, idx = lane & 15;
  float h = 0.f;
  for (int c = 0; c < NCHUNK / BATCH; ++c) {
    const int cid = b * (NCHUNK / BATCH) + c;
    if (half == 0) hs[(size_t)cid * DS + idx] = h;
    const float* M = Mc + (size_t)cid * 256;
    float hp = 0.f;
#pragma unroll
    for (int q = 0; q < 8; ++q) {
      const int k = 8 * half + q;
      hp += M[idx * DS + k] * __shfl(h, k, 32);
    }
    hp += __shfl_xor(hp, 16, 32);
    h = hp + bc[(size_t)cid * DS + idx];
  }
}

// ---------------------------------------------------------------------------
// Kernel 6: s[m] = g[m] . hs[chunk(m)] + partial[m]
// ---------------------------------------------------------------------------
__global__ __launch_bounds__(256) void compute_s_k(const float* __restrict__ g,
                                                   const float* __restrict__ part,
                                                   const float* __restrict__ hs,
                                                   float* __restrict__ sv) {
  const int m = blockIdx.x * 256 + threadIdx.x;
  if (m >= MTOT) return;
  const int cid = m >> 6;
  const float* gv = g + (size_t)m * DS;
  const float* hv = hs + (size_t)cid * DS;
  float a = part[m];
#pragma unroll
  for (int n = 0; n < DS; ++n) a += gv[n] * hv[n];
  sv[m] = a;
}

// ---------------------------------------------------------------------------
// Kernel 7: GEMM2  out[8192,1024] = (xc * silu(s + D*x_ssm)) @ Wout^T + x
// Prologue fused into A-tile load; B via async global->LDS; double-buffered;
// residual fused into epilogue.
// ---------------------------------------------------------------------------
__global__ __launch_bounds__(256) void gemm2_k(const float* __restrict__ Y1,
                                               const float* __restrict__ xc,
                                               const float* __restrict__ sv,
                                               const float* __restrict__ Dp,
                                               const float* __restrict__ Wout,
                                               const float* __restrict__ xres,
                                               float* __restrict__ out) {
  __shared__ float As[2][BLK][KP];
  __shared__ float Bs[2][BLK][KP];
  const int tid  = threadIdx.x;
  const int m0   = blockIdx.y * BLK;
  const int n0   = blockIdx.x * BLK;
  const int lane = tid & 31, wave = tid >> 5;
  const int half = lane >> 4, idx = lane & 15;
  const int wm   = (wave & 3) * 16;
  const int wn   = (wave >> 2) * 32;
  const int lr   = tid >> 2;
  const int c0   = (tid & 3) * 4;
  const float s_m = sv[m0 + lr];

  auto loadB = [&](int buf, int k0) {
    const float* src = Wout + (size_t)(n0 + lr) * DI + k0 + c0;
    async_load_b128(&Bs[buf][lr][c0],      src);
    async_load_b128(&Bs[buf][lr][c0 + 16], src + 16);
  };
  auto loadA = [&](int buf, int k0) {
    const float4 xv0 = *(const float4*)(xc + (size_t)(m0 + lr) * DI + k0 + c0);
    const float4 xv1 = *(const float4*)(xc + (size_t)(m0 + lr) * DI + k0 + c0 + 16);
    const float4 xs0 = *(const float4*)(Y1 + (size_t)(m0 + lr) * NPROJ + DI + k0 + c0);
    const float4 xs1 = *(const float4*)(Y1 + (size_t)(m0 + lr) * NPROJ + DI + k0 + c0 + 16);
    const float4 dp0 = *(const float4*)(Dp + k0 + c0);
    const float4 dp1 = *(const float4*)(Dp + k0 + c0 + 16);
    float4 a0, a1;
    float v;
    v = s_m + dp0.x * xs0.x; a0.x = xv0.x * silu_f(v);
    v = s_m + dp0.y * xs0.y; a0.y = xv0.y * silu_f(v);
    v = s_m + dp0.z * xs0.z; a0.z = xv0.z * silu_f(v);
    v = s_m + dp0.w * xs0.w; a0.w = xv0.w * silu_f(v);
    v = s_m + dp1.x * xs1.x; a1.x = xv1.x * silu_f(v);
    v = s_m + dp1.y * xs1.y; a1.y = xv1.y * silu_f(v);
    v = s_m + dp1.z * xs1.z; a1.z = xv1.z * silu_f(v);
    v = s_m + dp1.w * xs1.w; a1.w = xv1.w * silu_f(v);
    *(float4*)&As[buf][lr][c0]      = a0;
    *(float4*)&As[buf][lr][c0 + 16] = a1;
  };

  v8f acc0; v8f acc1;
#pragma unroll
  for (int q = 0; q < 8; ++q) { acc0[q] = 0.f; acc1[q] = 0.f; }

  loadB(0, 0);
  loadA(0, 0);
  wait_async0();
  __syncthreads();

  int buf = 0;
  for (int k0 = 0; k0 < DI; k0 += KC) {
    const int nb = buf ^ 1;
    const bool more = (k0 + KC < DI);
    if (more) loadB(nb, k0 + KC);
#pragma unroll
    for (int kk = 0; kk < KC; kk += 4) {
      const v2f a  = *(const v2f*)&As[buf][wm + idx][kk + 2 * half];
      const v2f b0 = *(const v2f*)&Bs[buf][wn + idx][kk + 2 * half];
      const v2f b1 = *(const v2f*)&Bs[buf][wn + 16 + idx][kk + 2 * half];
      acc0 = wmma4(a, b0, acc0);
      acc1 = wmma4(a, b1, acc1);
    }
    if (more) loadA(nb, k0 + KC);
    wait_async0();
    __syncthreads();
    buf = nb;
  }
#pragma unroll
  for (int r = 0; r < 8; ++r) {
    const int m = m0 + wm + 8 * half + r;
    const size_t o0 = (size_t)m * DM + n0 + wn + idx;
    const size_t o1 = o0 + 16;
    out[o0] = acc0[r] + xres[o0];
    out[o1] = acc1[r] + xres[o1];
  }
}

// ---------------------------------------------------------------------------
extern "C" void kernel_launch(void* const* d_in, const int* in_sizes, int n_in,
                              void* d_out, int out_size, void* d_ws, size_t ws_size,
                              hipStream_t stream) {
  (void)in_sizes; (void)n_in; (void)out_size; (void)ws_size;
  const float* x      = (const float*)d_in[0];
  const float* gamma  = (const float*)d_in[1];
  const float* beta   = (const float*)d_in[2];
  const float* w_in   = (const float*)d_in[3];   // [4096,1024]
  const float* conv_w = (const float*)d_in[4];   // [2048,1,4]
  const float* conv_b = (const float*)d_in[5];
  const float* A_par  = (const float*)d_in[6];   // [16,16]
  const float* D_par  = (const float*)d_in[7];
  const float* dt_w   = (const float*)d_in[8];   // [1,2048]
  const float* dt_b   = (const float*)d_in[9];   // [1]
  const float* B_w    = (const float*)d_in[10];  // [16,2048]
  const float* C_w    = (const float*)d_in[11];
  const float* w_out  = (const float*)d_in[12];  // [1024,2048]
  float* out = (float*)d_out;
  float* ws  = (float*)d_ws;

  size_t o = 0;
  float* Y1   = ws + o; o += (size_t)MTOT * NPROJ;   // 134 MB  x_proj
  float* XC   = ws + o; o += (size_t)MTOT * DI;      // 67 MB   conv+silu
  float* MU   = ws + o; o += MTOT;
  float* RS   = ws + o; o += MTOT;
  float* DT   = ws + o; o += MTOT;
  float* BMb  = ws + o; o += (size_t)MTOT * DS;
  float* CMb  = ws + o; o += (size_t)MTOT * DS;
  float* G    = ws + o; o += (size_t)MTOT * DS;
  float* PART = ws + o; o += MTOT;
  float* SV   = ws + o; o += MTOT;
  float* MC   = ws + o; o += (size_t)NCHUNK * 256;
  float* BC   = ws + o; o += (size_t)NCHUNK * DS;
  float* HS   = ws + o; o += (size_t)NCHUNK * DS;

  ln_stats_k<<<MTOT, 256, 0, stream>>>(x, MU, RS);
  gemm1_ln_k<<<dim3(NPROJ / BLK, MTOT / BLK), 256, 0, stream>>>(x, MU, RS, gamma,
                                                                beta, w_in, Y1);
  conv_proj_k<<<MTOT, 256, 0, stream>>>(Y1, conv_w, conv_b, dt_w, dt_b, B_w, C_w,
                                        XC, DT, BMb, CMb);
  scan_chunks_k<<<NCHUNK, 32, 0, stream>>>(DT, BMb, CMb, A_par, G, PART, MC, BC);
  combine_k<<<1, 64, 0, stream>>>(MC, BC, HS);
  compute_s_k<<<MTOT / 256, 256, 0, stream>>>(G, PART, HS, SV);
  gemm2_k<<<dim3(DM / BLK, MTOT / BLK), 256, 0, stream>>>(Y1, XC, SV, D_par, w_out,
                                                          x, out);
}